// DANet_21320217657761
// MI455X (gfx1250) — compile-verified
//
#include <hip/hip_runtime.h>
#include <hip/hip_bf16.h>
#include <math.h>

// ---------------------------------------------------------------------------
// DANet soft-kmeans for MI455X (gfx1250).
//  B=16, D=40, N=64512, K=4, 10 iterations, alpha=5.
//  Pipeline: k_tanh -> k_init -> 10x (k_accum -> k_update) -> k_mask
//  Vact (165MB) is L2-resident (192MB L2), so iteration kernels stream L2.
//  cross = U x Vact uses V_WMMA_F32_16X16X4_F32 (K padded 4->16 with zeros).
// ---------------------------------------------------------------------------

#define BB      16
#define DD      40
#define NPTS    64512
#define KCL     4
#define CHUNK   256
#define WGPB    63          // workgroups per batch: 63 * 4 * 256 = 64512
#define CPW     4           // chunks per workgroup
#define ALPHA_C 5.0f
#define VSTRIDE 257         // LDS tile row stride (odd -> conflict-free columns)
#define USTRIDE 41          // padded-U row stride

typedef float vf2 __attribute__((ext_vector_type(2)));
typedef float vf8 __attribute__((ext_vector_type(8)));

#if defined(__gfx1250__) && __has_builtin(__builtin_amdgcn_wmma_f32_16x16x4_f32)
#define WMMA_F32X4_OK 1
#else
#define WMMA_F32X4_OK 0
#endif

// D(16x16,f32) = A(16x4,f32) * B(4x16,f32) + C
// A layout: lanes 0-15 hold (M=lane, K=0/1) in vgpr0/1; lanes 16-31 K=2/3.
// B layout assumed: K = 2*(lane>=16) + vgpr, col = lane&15.
// C layout: vgpr r, lanes 0-15 -> M=r, col=lane&15 (ISA 7.12.2).
__device__ __forceinline__ vf8 wmma_f32x4(vf2 a, vf2 b, vf8 c) {
#if WMMA_F32X4_OK
  return __builtin_amdgcn_wmma_f32_16x16x4_f32(false, a, false, b, (short)0, c,
                                               false, false);
#else
  // placeholder so host pass / non-gfx1250 still parses; never used on target
  c[0] = fmaf(a[0], b[0], c[0]);
  c[1] = fmaf(a[1], b[1], c[1]);
  return c;
#endif
}

// Raw-rate transcendentals: single v_sqrt_f32 / v_rcp_f32 (~1 ulp), skipping
// the precise-IEEE correction sequences libm inserts.  Softmax-tolerant.
__device__ __forceinline__ float fast_sqrt(float x) {
#if defined(__HIP_DEVICE_COMPILE__)
  return __builtin_amdgcn_sqrtf(x);
#else
  return sqrtf(x);
#endif
}
__device__ __forceinline__ float fast_rcp(float x) {
#if defined(__HIP_DEVICE_COMPILE__)
  return __builtin_amdgcn_rcpf(x);
#else
  return 1.0f / x;
#endif
}

// ---------------------------------------------------------------------------
// Kernel 1: Vact = tanh(V), vectorized float4.  41,287,680 elems = 10,321,920 f4
// ---------------------------------------------------------------------------
__global__ __launch_bounds__(256) void k_tanh(const float4* __restrict__ v,
                                              float4* __restrict__ o, int n4) {
  int i = blockIdx.x * 256 + threadIdx.x;
  if (i < n4) {
    float4 x = v[i];
    float4 y;
    y.x = tanhf(x.x); y.y = tanhf(x.y); y.z = tanhf(x.z); y.w = tanhf(x.w);
    o[i] = y;
  }
}

// ---------------------------------------------------------------------------
// Kernel 2: U0[b,k,d] = Vact[b,d,idx[b,k]]; zero the accumulators.
// ---------------------------------------------------------------------------
__global__ __launch_bounds__(192) void k_init(const float* __restrict__ vact,
                                              const long long* __restrict__ idx,
                                              float* __restrict__ U,
                                              float* __restrict__ VY,
                                              float* __restrict__ sumY) {
  const int b = blockIdx.x;
  const int t = threadIdx.x;
  if (t < KCL * DD) {
    const int k = t / DD, d = t % DD;
    const int n = (int)idx[b * KCL + k];
    U[b * KCL * DD + t] = vact[(size_t)b * DD * NPTS + (size_t)d * NPTS + n];
    VY[b * KCL * DD + t] = 0.0f;
  }
  if (t < KCL) sumY[b * KCL + t] = 0.0f;
}

// ---------------------------------------------------------------------------
// Kernel 3: one k-means accumulation pass.
//  grid = (WGPB, B), block = 256 (8 wave32).  Each WG owns 4 chunks of 256 pts.
//  Per chunk: stage tile in LDS -> WMMA cross -> softmax -> Y tile in LDS ->
//  (k,d)-owner threads accumulate V_Y over points -> global f32 atomics once.
// ---------------------------------------------------------------------------
__global__ __launch_bounds__(256) void k_accum(const float* __restrict__ vact,
                                               const float* __restrict__ U,
                                               float* __restrict__ VY,
                                               float* __restrict__ sumY) {
  __shared__ float Upad[16 * USTRIDE];      // 16x40 padded (rows 4..15 = 0)
  __shared__ float u2_s[16];
  __shared__ float Vl[DD * VSTRIDE];        // 40 x 256 tile, stride 257
  __shared__ float v2_l[CHUNK];
  __shared__ float Yt[KCL * CHUNK];         // Y transposed: [k][p]

  const int b    = blockIdx.y;
  const int t    = threadIdx.x;
  const int lane = t & 31;
  const int wave = t >> 5;
  const int m    = lane & 15;               // column / M-row selector
  const int h    = lane >> 4;               // half-wave selector
  const float* vb = vact + (size_t)b * DD * NPTS;

  // stage padded U into LDS
  for (int i = t; i < 16 * DD; i += 256) {
    const int r = i / DD, c = i % DD;
    Upad[r * USTRIDE + c] = (r < KCL) ? U[b * KCL * DD + r * DD + c] : 0.0f;
  }
  __syncthreads();
  if (t < 16) {
    float s = 0.0f;
    for (int d = 0; d < DD; ++d) { float x = Upad[t * USTRIDE + d]; s += x * x; }
    u2_s[t] = s;
  }
  __syncthreads();
  const float u2r0 = u2_s[0], u2r1 = u2_s[1], u2r2 = u2_s[2], u2r3 = u2_s[3];

  // preload all 10 A-fragments of U (wave-invariant for the whole kernel)
  vf2 af[10];
#pragma unroll
  for (int j = 0; j < 10; ++j) {
    const int c0 = 4 * j + 2 * h;
    af[j].x = Upad[m * USTRIDE + c0];
    af[j].y = Upad[m * USTRIDE + c0 + 1];
  }

  // (k,d)-owner accumulators, carried in registers across all 4 chunks
  float acc = 0.0f, sacc = 0.0f;
  const int od = t % DD, ok = t / DD;       // valid for t < 160
  const int sk = t - 160;                   // valid for 160 <= t < 164

  const int wgbase = blockIdx.x * (CPW * CHUNK);

  for (int cc = 0; cc < CPW; ++cc) {
    const int n0c = wgbase + cc * CHUNK;
    __syncthreads();                        // protect Vl/Yt reuse

    // --- stage tile (coalesced over points) and fold in v2 ---
    {
      const int p = t;                      // 256 threads == CHUNK
      const float* src = vb + n0c + p;
      float v2 = 0.0f;
#pragma unroll 8
      for (int d = 0; d < DD; ++d) {
        const float x = src[(size_t)d * NPTS];
        Vl[d * VSTRIDE + p] = x;
        v2 += x * x;
      }
      v2_l[p] = v2;
      if (cc + 1 < CPW)                     // hint next chunk into caches
        __builtin_prefetch(vb + (size_t)(t % DD) * NPTS + n0c + CHUNK +
                               (t / DD) * 40, 0, 1);
    }
    __syncthreads();

    // --- cross = U x Vtile via WMMA, then per-point softmax over K=4 ---
    for (int pass = 0; pass < 2; ++pass) {
      const int p0 = ((wave << 1) | pass) << 4;   // 16-point subtile base
      vf8 c = {0.f, 0.f, 0.f, 0.f, 0.f, 0.f, 0.f, 0.f};
#pragma unroll
      for (int j = 0; j < 10; ++j) {
        const int dr = 4 * j + 2 * h;
        vf2 bf;
        bf.x = Vl[dr * VSTRIDE + p0 + m];
        bf.y = Vl[(dr + 1) * VSTRIDE + p0 + m];
        c = wmma_f32x4(af[j], bf, c);
      }
      // lanes 0..15: c[0..3] = cross for clusters 0..3 at point p0+m
      const float v2p = v2_l[p0 + m];
      float d0 = -ALPHA_C * fast_sqrt(fmaxf(v2p + u2r0 - 2.0f * c[0], 1e-12f));
      float d1 = -ALPHA_C * fast_sqrt(fmaxf(v2p + u2r1 - 2.0f * c[1], 1e-12f));
      float d2 = -ALPHA_C * fast_sqrt(fmaxf(v2p + u2r2 - 2.0f * c[2], 1e-12f));
      float d3 = -ALPHA_C * fast_sqrt(fmaxf(v2p + u2r3 - 2.0f * c[3], 1e-12f));
      const float mx = fmaxf(fmaxf(d0, d1), fmaxf(d2, d3));
      const float e0 = __expf(d0 - mx), e1 = __expf(d1 - mx);
      const float e2 = __expf(d2 - mx), e3 = __expf(d3 - mx);
      const float inv = fast_rcp(e0 + e1 + e2 + e3);
      if (h == 0) {
        Yt[0 * CHUNK + p0 + m] = e0 * inv;
        Yt[1 * CHUNK + p0 + m] = e1 * inv;
        Yt[2 * CHUNK + p0 + m] = e2 * inv;
        Yt[3 * CHUNK + p0 + m] = e3 * inv;
      }
    }
    __syncthreads();

    // --- V_Y[k,d] += sum_p Y[k,p] * V[d,p] ; sum_Y[k] += sum_p Y[k,p] ---
    if (t < KCL * DD) {
      const float* yr = &Yt[ok * CHUNK];
      const float* vr = &Vl[od * VSTRIDE];
#pragma unroll 8
      for (int p = 0; p < CHUNK; ++p) acc = fmaf(yr[p], vr[p], acc);
    } else if (t < KCL * DD + KCL) {
      const float* yr = &Yt[sk * CHUNK];
#pragma unroll 8
      for (int p = 0; p < CHUNK; ++p) sacc += yr[p];
    }
  }

  if (t < KCL * DD)            atomicAdd(&VY[b * KCL * DD + ok * DD + od], acc);
  else if (t < KCL * DD + KCL) atomicAdd(&sumY[b * KCL + sk], sacc);
}

// ---------------------------------------------------------------------------
// Kernel 4: U = V_Y / (sum_Y + eps); re-zero accumulators; optionally emit A.
// ---------------------------------------------------------------------------
__global__ __launch_bounds__(192) void k_update(float* __restrict__ U,
                                                float* __restrict__ VY,
                                                float* __restrict__ sumY,
                                                float* __restrict__ outA) {
  const int b = blockIdx.x;
  const int t = threadIdx.x;
  if (t < KCL * DD) {
    const int k = t / DD;
    const float u = VY[b * KCL * DD + t] / (sumY[b * KCL + k] + 1e-9f);
    U[b * KCL * DD + t] = u;
    if (outA) outA[b * KCL * DD + t] = u;
  }
  __syncthreads();
  if (t < KCL * DD) VY[b * KCL * DD + t] = 0.0f;
  if (t < KCL)      sumY[b * KCL + t] = 0.0f;
}

// ---------------------------------------------------------------------------
// Kernel 5: mask = softmax_k(A x Vact).  WMMA dot-product, B-frags straight
// from global (coalesced 64B half-wave rows), stores coalesced per cluster row.
// ---------------------------------------------------------------------------
__global__ __launch_bounds__(256) void k_mask(const float* __restrict__ vact,
                                              const float* __restrict__ U,
                                              float* __restrict__ mask) {
  __shared__ float Upad[16 * USTRIDE];

  const int b    = blockIdx.y;
  const int t    = threadIdx.x;
  const int lane = t & 31;
  const int wave = t >> 5;
  const int m    = lane & 15;
  const int h    = lane >> 4;
  const float* vb = vact + (size_t)b * DD * NPTS;
  float* mb = mask + (size_t)b * KCL * NPTS;

  for (int i = t; i < 16 * DD; i += 256) {
    const int r = i / DD, c = i % DD;
    Upad[r * USTRIDE + c] = (r < KCL) ? U[b * KCL * DD + r * DD + c] : 0.0f;
  }
  __syncthreads();

  vf2 af[10];
#pragma unroll
  for (int j = 0; j < 10; ++j) {
    const int c0 = 4 * j + 2 * h;
    af[j].x = Upad[m * USTRIDE + c0];
    af[j].y = Upad[m * USTRIDE + c0 + 1];
  }

  const int wgbase = blockIdx.x * (CPW * CHUNK);   // 1024 points per WG
  for (int g = 0; g < 8; ++g) {                    // 8 waves x 8 subtiles x 16
    const int n0 = wgbase + (wave + 8 * g) * 16;
    vf8 c = {0.f, 0.f, 0.f, 0.f, 0.f, 0.f, 0.f, 0.f};
#pragma unroll
    for (int j = 0; j < 10; ++j) {
      const int dr = 4 * j + 2 * h;
      vf2 bf;
      bf.x = vb[(size_t)dr * NPTS + n0 + m];
      bf.y = vb[(size_t)(dr + 1) * NPTS + n0 + m];
      c = wmma_f32x4(af[j], bf, c);
    }
    const float mx = fmaxf(fmaxf(c[0], c[1]), fmaxf(c[2], c[3]));
    const float e0 = __expf(c[0] - mx), e1 = __expf(c[1] - mx);
    const float e2 = __expf(c[2] - mx), e3 = __expf(c[3] - mx);
    const float inv = fast_rcp(e0 + e1 + e2 + e3);
    if (h == 0) {
      mb[0 * (size_t)NPTS + n0 + m] = e0 * inv;
      mb[1 * (size_t)NPTS + n0 + m] = e1 * inv;
      mb[2 * (size_t)NPTS + n0 + m] = e2 * inv;
      mb[3 * (size_t)NPTS + n0 + m] = e3 * inv;
    }
  }
}

// ---------------------------------------------------------------------------
extern "C" void kernel_launch(void* const* d_in, const int* in_sizes, int n_in,
                              void* d_out, int out_size, void* d_ws,
                              size_t ws_size, hipStream_t stream) {
  (void)in_sizes; (void)n_in; (void)out_size; (void)ws_size;

  const float*      V   = (const float*)d_in[0];
  const long long*  idx = (const long long*)d_in[1];   // jnp.int64

  float* out     = (float*)d_out;
  float* outMask = out;                                        // (B,K,N)
  float* outVact = out + (size_t)BB * KCL * NPTS;              // (B,D,N)
  float* outA    = outVact + (size_t)BB * DD * NPTS;           // (B,K,D)

  float* ws   = (float*)d_ws;
  float* U    = ws;                                            // B*K*D
  float* VY   = ws + BB * KCL * DD;                            // B*K*D
  float* sumY = VY + BB * KCL * DD;                            // B*K

  const int n4 = (BB * DD * NPTS) / 4;                         // 10,321,920
  k_tanh<<<n4 / 256, 256, 0, stream>>>((const float4*)V, (float4*)outVact, n4);

  k_init<<<BB, 192, 0, stream>>>(outVact, idx, U, VY, sumY);

  const dim3 gwide(WGPB, BB);
  for (int it = 0; it < 10; ++it) {
    k_accum<<<gwide, 256, 0, stream>>>(outVact, U, VY, sumY);
    k_update<<<BB, 192, 0, stream>>>(U, VY, sumY, (it == 9) ? outA : nullptr);
  }

  k_mask<<<gwide, 256, 0, stream>>>(outVact, U, outMask);
}